// Loss_46102178955400
// MI455X (gfx1250) — compile-verified
//
#include <hip/hip_runtime.h>
#include <hip/hip_bf16.h>
#include <math.h>

typedef __attribute__((ext_vector_type(2))) float v2f;
typedef __attribute__((ext_vector_type(8))) float v8f;

#define GK 64          // members per group (K in reference)
#define CT 32          // columns per LDS chunk
#define TP (CT + 4)    // padded row stride (floats) -> 144 B, keeps 16 B alignment

// Generic pointer to __shared__ -> LDS byte offset (aperture maps addr[31:0] to LDS).
__device__ __forceinline__ unsigned lds_off(const void* p) {
    return (unsigned)(uintptr_t)p;
}

// ---------------- Kernel 0: zero accumulators ----------------
__global__ void init_acc_kernel(float* acc) {
    if (threadIdx.x < 3) acc[threadIdx.x] = 0.0f;
}

// ---------------- Kernel 1: cross-entropy ----------------
// One wave (32 lanes) per row. blockDim = 256 -> 8 rows per block.
__global__ __launch_bounds__(256) void ce_kernel(
    const float* __restrict__ logits, const int* __restrict__ targets,
    float* __restrict__ acc, int nrows, int C) {
    const int wave = threadIdx.x >> 5;
    const int lane = threadIdx.x & 31;
    const int row  = blockIdx.x * 8 + wave;
    if (row >= nrows) return;

    const float* r = logits + (size_t)row * C;
    float m = -INFINITY;
    for (int c = lane; c < C; c += 32) m = fmaxf(m, r[c]);
    #pragma unroll
    for (int off = 16; off > 0; off >>= 1) m = fmaxf(m, __shfl_xor(m, off, 32));

    float s = 0.0f;
    for (int c = lane; c < C; c += 32) s += expf(r[c] - m);
    #pragma unroll
    for (int off = 16; off > 0; off >>= 1) s += __shfl_xor(s, off, 32);

    if (lane == 0) {
        const int t = targets[row];
        const float logp = r[t] - m - logf(s);
        atomicAdd(acc, -logp);
    }
}

// ---------------- Kernel 2: per-group alignment + Gram + robustness ----------------
// grid = G, block = 256 (8 waves). Gram = members @ members^T via exact-f32
// V_WMMA_F32_16X16X4_F32, 4x4 grid of 16x16 tiles, 2 tiles/wave (same tile column
// -> shared B operand). Member tiles are double-buffered into LDS with
// GLOBAL_LOAD_ASYNC_TO_LDS_B128 (ASYNCcnt) to overlap the gather with WMMA.
__global__ __launch_bounds__(256) void group_kernel(
    const float* __restrict__ train_logits,
    const float* __restrict__ unl,
    const int*   __restrict__ centroid_ids,
    const int*   __restrict__ member_ids,
    float* __restrict__ align_acc,
    float* __restrict__ robust_acc,
    int C) {
    __shared__ __align__(16) float tile[2][GK][TP];   // double-buffered member chunk
    __shared__ float centb[2][CT];                    // double-buffered centroid chunk
    __shared__ int   mids[GK];
    __shared__ float gram_s[GK][GK + 1];              // 64x64 Gram
    __shared__ float red[256];

    const int g    = blockIdx.x;
    const int tid  = threadIdx.x;
    const int lane = tid & 31;
    const int wave = tid >> 5;

    if (tid < GK) mids[tid] = member_ids[g * GK + tid];
    __syncthreads();

    const int cid = centroid_ids[g];
    const float* crow = train_logits + (size_t)cid * C;

    // Wave owns tiles (tm0, tn) and (tm1 = tm0+2, tn): same tile column -> shared B.
    const int tm0 = wave >> 2, tn = wave & 3, tm1 = tm0 + 2;
    const int l15 = lane & 15;        // matrix row (A) / col (B) within tile
    const int khalf = lane >> 4;      // selects K pair {0,1} vs {2,3}
    v8f acc0 = {}; v8f acc1 = {};

    // Alignment: thread handles member (tid>>2), column residue (tid&3).
    const int am = tid >> 2, asub = tid & 3;
    float apart = 0.0f;

    // Async copy geometry: 2048 floats/chunk = 256 thr * 2 pieces * float4.
    const int prow0 = tid >> 3;            // rows 0..31
    const int prow1 = 32 + (tid >> 3);     // rows 32..63
    const int pcol  = (tid & 7) * 4;       // 0,4,...,28 (16 B aligned)
    const float* gbase0 = unl + (size_t)mids[prow0] * C + pcol;
    const float* gbase1 = unl + (size_t)mids[prow1] * C + pcol;

    const int nFull = C / CT;
    const int CFULL = nFull * CT;

    auto issue_async = [&](int buf, int c0) {
        const unsigned l0 = lds_off(&tile[buf][prow0][pcol]);
        const unsigned l1 = lds_off(&tile[buf][prow1][pcol]);
        const float* g0 = gbase0 + c0;
        const float* g1 = gbase1 + c0;
        asm volatile("global_load_async_to_lds_b128 %0, %1, off"
                     :: "v"(l0), "v"(g0) : "memory");
        asm volatile("global_load_async_to_lds_b128 %0, %1, off"
                     :: "v"(l1), "v"(g1) : "memory");
    };

    auto compute_chunk = [&](int buf) {
        #pragma unroll
        for (int kk = 0; kk < CT; kk += 4) {
            const int k0 = kk + 2 * khalf;          // ISA f32 16x16x4 A/B layout
            v2f a0, b0, a1;
            a0.x = tile[buf][tm0 * 16 + l15][k0]; a0.y = tile[buf][tm0 * 16 + l15][k0 + 1];
            b0.x = tile[buf][tn  * 16 + l15][k0]; b0.y = tile[buf][tn  * 16 + l15][k0 + 1];
            a1.x = tile[buf][tm1 * 16 + l15][k0]; a1.y = tile[buf][tm1 * 16 + l15][k0 + 1];
            acc0 = __builtin_amdgcn_wmma_f32_16x16x4_f32(
                false, a0, false, b0, (short)0, acc0, false, false);
            acc1 = __builtin_amdgcn_wmma_f32_16x16x4_f32(
                false, a1, false, b0, (short)0, acc1, false, false);
        }
        #pragma unroll
        for (int c = asub; c < CT; c += 4) {
            const float d = tile[buf][am][c] - centb[buf][c];
            apart += d * d;
        }
    };

    int buf = 0;
    if (nFull > 0) {
        // Prologue: chunk 0 into buffer 0.
        if (tid < CT) centb[0][tid] = crow[tid];
        issue_async(0, 0);
        asm volatile("s_wait_asynccnt 0x0" ::: "memory");
        __syncthreads();

        for (int ch = 0; ch < nFull; ++ch) {
            if (ch + 1 < nFull) {
                if (tid < CT) centb[buf ^ 1][tid] = crow[(ch + 1) * CT + tid];
                issue_async(buf ^ 1, (ch + 1) * CT);   // overlaps with compute below
            }
            compute_chunk(buf);
            asm volatile("s_wait_asynccnt 0x0" ::: "memory");
            __syncthreads();
            buf ^= 1;
        }
    }

    // Ragged tail (cols [CFULL, C)): guarded synchronous fill, zero-padded.
    if (CFULL < C) {
        for (int i = tid; i < GK * CT; i += 256) {
            const int r = i >> 5;              // CT == 32
            const int c = i & (CT - 1);
            const int gc = CFULL + c;
            tile[buf][r][c] = (gc < C) ? unl[(size_t)mids[r] * C + gc] : 0.0f;
        }
        if (tid < CT) {
            const int gc = CFULL + tid;
            centb[buf][tid] = (gc < C) ? crow[gc] : 0.0f;
        }
        __syncthreads();
        compute_chunk(buf);
        __syncthreads();
    }

    // Spill Gram accumulators per C/D layout:
    // VGPR r -> M = 16*tm + r + 8*khalf, N = 16*tn + (lane&15).
    #pragma unroll
    for (int r = 0; r < 8; ++r) {
        gram_s[tm0 * 16 + r + 8 * khalf][tn * 16 + l15] = acc0[r];
        gram_s[tm1 * 16 + r + 8 * khalf][tn * 16 + l15] = acc1[r];
    }
    red[tid] = apart;
    __syncthreads();

    // Alignment: per-member sqrt of total, then block sum.
    float aval = 0.0f;
    if (tid < GK) {
        const float s = red[4 * tid] + red[4 * tid + 1] + red[4 * tid + 2] + red[4 * tid + 3];
        aval = sqrtf(s);
    }
    // Robustness: upper-triangle pairwise distances from Gram (safe sqrt).
    float rpart = 0.0f;
    for (int idx = tid; idx < GK * GK; idx += 256) {
        const int i = idx >> 6, j = idx & 63;
        if (j > i) {
            const float d2 = gram_s[i][i] + gram_s[j][j] - 2.0f * gram_s[i][j];
            rpart += (d2 > 0.0f) ? sqrtf(d2) : 0.0f;
        }
    }
    __syncthreads();

    red[tid] = aval;
    #pragma unroll
    for (int s = 128; s > 0; s >>= 1) {
        __syncthreads();
        if (tid < s) red[tid] += red[tid + s];
    }
    __syncthreads();
    if (tid == 0) atomicAdd(align_acc, red[0]);
    __syncthreads();

    red[tid] = rpart;
    #pragma unroll
    for (int s = 128; s > 0; s >>= 1) {
        __syncthreads();
        if (tid < s) red[tid] += red[tid + s];
    }
    __syncthreads();
    if (tid == 0) atomicAdd(robust_acc, red[0]);
}

// ---------------- Kernel 3: combine ----------------
__global__ void finalize_kernel(const float* __restrict__ acc, float* __restrict__ out,
                                float ceScale, float alignScale, float robustScale) {
    out[0] = acc[0] * ceScale + acc[1] * alignScale + acc[2] * robustScale;
}

extern "C" void kernel_launch(void* const* d_in, const int* in_sizes, int n_in,
                              void* d_out, int out_size, void* d_ws, size_t ws_size,
                              hipStream_t stream) {
    const float* train_logits  = (const float*)d_in[0];
    const int*   train_targets = (const int*)  d_in[1];
    const float* unl           = (const float*)d_in[2];
    const int*   centroid_ids  = (const int*)  d_in[3];
    const int*   member_ids    = (const int*)  d_in[4];

    const int N_TRAIN = in_sizes[1];                 // 4096
    const int C       = in_sizes[0] / N_TRAIN;       // 1000
    const int N_UNL   = in_sizes[2] / C;             // 32768
    const int G       = in_sizes[3];                 // 512
    const int K       = in_sizes[4] / G;             // 64 (== GK)

    float* acc = (float*)d_ws;   // acc[0]=ce sum, acc[1]=align sum, acc[2]=robust sum

    init_acc_kernel<<<1, 32, 0, stream>>>(acc);
    ce_kernel<<<(N_TRAIN + 7) / 8, 256, 0, stream>>>(train_logits, train_targets,
                                                     acc + 0, N_TRAIN, C);
    group_kernel<<<G, 256, 0, stream>>>(train_logits, unl, centroid_ids, member_ids,
                                        acc + 1, acc + 2, C);
    const float ceScale     = 1.0f / (float)N_TRAIN;
    const float alignScale  = 1.0f / (float)N_UNL;                    // LAMBDA_1 = 1.0
    const float robustScale = 0.5f / ((float)K * (float)N_UNL);       // LAMBDA_2 = 0.5
    finalize_kernel<<<1, 1, 0, stream>>>(acc, (float*)d_out, ceScale, alignScale, robustScale);
}